// myModel_32667521253780
// MI455X (gfx1250) — compile-verified
//
#include <hip/hip_runtime.h>
#include <stdint.h>

// Problem constants (fixed by the reference).
#define B_TOT   262144
#define NNUM    64
#define FCAT    32
#define NCAT    8
#define NPHEN   96        // N_NUM + F_CAT
#define NFEAT   320       // N_NUM + F_CAT*N_CAT
#define BLK     128       // 4 waves (wave32)
#define ROWD    68        // dwords per LDS row: 64 data + 4 pad -> 272B (16B aligned, 2-cycle b128)

__device__ __forceinline__ float fast_tanhf(float x) {
#if __has_builtin(__builtin_amdgcn_tanhf)
  return __builtin_amdgcn_tanhf(x);
#else
  float r;
  asm("v_tanh_f32 %0, %1" : "=v"(r) : "v"(x));
  return r;
#endif
}

__global__ __launch_bounds__(BLK, 1) void pheno_softmax_kernel(
    const float* __restrict__ xnum,   // (B, 64)
    const float* __restrict__ xcat,   // (32, B, 8)
    const float* __restrict__ W,      // (2, 320)
    const float* __restrict__ M,      // (2, 96)
    float* __restrict__ out)          // (2, B)
{
  __shared__ float sX[BLK * ROWD];
  const int tid = threadIdx.x;
  const int b0  = blockIdx.x * BLK;
  const int b   = b0 + tid;

  // ---- Stage 128 x 64 f32 numerical rows into LDS via CDNA5 async DMA.
  // Global side: each issue moves a fully-coalesced 512B/wave (lane-contiguous
  // float4s). LDS side: rows padded to 272B so later ds_load_b128 reads are
  // 16B-aligned and bank-conflict-optimal (2 cycles, the b128 minimum).
  {
    const float* gsrc = xnum + (size_t)b0 * NNUM;   // 32KB contiguous
    unsigned ldsbase = (unsigned)(uintptr_t)(void*)sX; // low 32b of flat LDS addr = LDS offset
#pragma unroll
    for (int i = 0; i < 16; ++i) {
      int g    = i * BLK + tid;       // float4 index within the 128x16-float4 tile
      int row  = g >> 4;
      int col4 = g & 15;
      unsigned ldsaddr = ldsbase + (unsigned)(row * (ROWD * 4) + col4 * 16);
      int goff = g * 16;              // byte offset into gsrc (lane-contiguous)
      asm volatile("global_load_async_to_lds_b128 %0, %1, %2"
                   :: "v"(ldsaddr), "v"(goff), "s"(gsrc)
                   : "memory");
    }
  }
#if __has_builtin(__builtin_amdgcn_s_wait_asynccnt)
  __builtin_amdgcn_s_wait_asynccnt(0);
#else
  asm volatile("s_wait_asynccnt 0" ::: "memory");
#endif
  __syncthreads();

  float f0 = 0.0f, f1 = 0.0f;

  // ---- Numerical part: f_c += sum_i M[c,i] * tanh(W[c,i] * x[b,i])
  // W/M indices are compile-time constants -> uniform -> s_load (K$ path).
  const float* xr = &sX[tid * ROWD];
#pragma unroll
  for (int i = 0; i < NNUM; i += 4) {
    float4 x = *(const float4*)(xr + i);   // ds_load_b128 from LDS
    f0 = fmaf(M[i + 0],         fast_tanhf(W[i + 0]         * x.x), f0);
    f1 = fmaf(M[NPHEN + i + 0], fast_tanhf(W[NFEAT + i + 0] * x.x), f1);
    f0 = fmaf(M[i + 1],         fast_tanhf(W[i + 1]         * x.y), f0);
    f1 = fmaf(M[NPHEN + i + 1], fast_tanhf(W[NFEAT + i + 1] * x.y), f1);
    f0 = fmaf(M[i + 2],         fast_tanhf(W[i + 2]         * x.z), f0);
    f1 = fmaf(M[NPHEN + i + 2], fast_tanhf(W[NFEAT + i + 2] * x.z), f1);
    f0 = fmaf(M[i + 3],         fast_tanhf(W[i + 3]         * x.w), f0);
    f1 = fmaf(M[NPHEN + i + 3], fast_tanhf(W[NFEAT + i + 3] * x.w), f1);
  }

  // ---- Categorical part: f_c += sum_f M[c,64+f] * tanh(W_cat[c,f,:]·xcat[f,b,:])
  // xcat[f,b,:] is 32B contiguous; consecutive lanes hit consecutive 32B
  // chunks -> global_load_b128s are naturally fully coalesced.
#pragma unroll
  for (int f = 0; f < FCAT; ++f) {
    const float* cp = xcat + ((size_t)f * B_TOT + b) * NCAT;
    float4 a = *(const float4*)cp;
    float4 c = *(const float4*)(cp + 4);
    const int w0 = NNUM + f * NCAT;           // W[0, 64 + 8f + n]
    const int w1 = NFEAT + w0;                // W[1, 64 + 8f + n]
    float d0 = W[w0+0]*a.x; float d1 = W[w1+0]*a.x;
    d0 = fmaf(W[w0+1], a.y, d0);  d1 = fmaf(W[w1+1], a.y, d1);
    d0 = fmaf(W[w0+2], a.z, d0);  d1 = fmaf(W[w1+2], a.z, d1);
    d0 = fmaf(W[w0+3], a.w, d0);  d1 = fmaf(W[w1+3], a.w, d1);
    d0 = fmaf(W[w0+4], c.x, d0);  d1 = fmaf(W[w1+4], c.x, d1);
    d0 = fmaf(W[w0+5], c.y, d0);  d1 = fmaf(W[w1+5], c.y, d1);
    d0 = fmaf(W[w0+6], c.z, d0);  d1 = fmaf(W[w1+6], c.z, d1);
    d0 = fmaf(W[w0+7], c.w, d0);  d1 = fmaf(W[w1+7], c.w, d1);
    f0 = fmaf(M[NNUM + f],         fast_tanhf(d0), f0);
    f1 = fmaf(M[NPHEN + NNUM + f], fast_tanhf(d1), f1);
  }

  // ---- 2-class softmax == sigmoid: p0 = 1/(1+e^{f1-f0}), p1 = e*p0
  float e  = __expf(f1 - f0);     // v_exp_f32
  float p0 = 1.0f / (1.0f + e);
  out[b]         = p0;
  out[B_TOT + b] = e * p0;
}

extern "C" void kernel_launch(void* const* d_in, const int* in_sizes, int n_in,
                              void* d_out, int out_size, void* d_ws, size_t ws_size,
                              hipStream_t stream) {
  (void)in_sizes; (void)n_in; (void)out_size; (void)d_ws; (void)ws_size;
  const float* xnum = (const float*)d_in[0];
  const float* xcat = (const float*)d_in[1];
  const float* W    = (const float*)d_in[2];
  const float* M    = (const float*)d_in[3];
  float* out        = (float*)d_out;

  dim3 grid(B_TOT / BLK);
  dim3 block(BLK);
  hipLaunchKernelGGL(pheno_softmax_kernel, grid, block, 0, stream,
                     xnum, xcat, W, M, out);
}